// TokenEmbedding_60464549593464
// MI455X (gfx1250) — compile-verified
//
#include <hip/hip_runtime.h>

typedef __attribute__((ext_vector_type(2))) float v2f;
typedef __attribute__((ext_vector_type(8))) float v8f;

#define TAO      3
#define MDEL     5
#define C_IN     7
#define KERNELS  73
#define NFEAT    18              // (M+1)*KSIZE
#define BB       32
#define LL       4096
#define DMODEL   512
#define NTILE    5               // 5*16 = 80 >= 74 channels per input-channel group
#define TPW      8               // position tiles per wave
#define WPB      8               // waves per block (256 threads, wave32)
#define XBLOCKS  (LL / 16 / (TPW * WPB))   // 4

// Fused delay-embedding + circular conv as per-channel GEMM on the f32 WMMA path.
// Wave = 16 positions x 16 channels tile; K=18 features padded to 20 -> 5x
// v_wmma_f32_16x16x4_f32 per tile. Leftout channel folded as column 73 of c==6.
// Interior tiles take a fast path: per-lane base pointer + constant byte offsets.
__global__ __launch_bounds__(256)
void delay_conv_wmma_kernel(const float* __restrict__ x,
                            const float* __restrict__ conv_w,
                            const float* __restrict__ conv_b,
                            const float* __restrict__ leftout_w,
                            const float* __restrict__ leftout_b,
                            float* __restrict__ out) {
    const int lane = threadIdx.x & 31;
    const int wv   = threadIdx.x >> 5;

    // scalar decode: no per-lane division chains
    const int b     = blockIdx.z;
    const int c     = blockIdx.y / NTILE;
    const int ntile = blockIdx.y - c * NTILE;
    const int l0_base = (blockIdx.x * WPB + wv) * (TPW * 16);

    const int half = lane >> 4;    // K-half selector (A/B frag layout)
    const int nloc = lane & 15;    // N within tile
    const int row  = lane & 15;    // M within tile
    const int n    = ntile * 16 + nloc;
    const int Nc   = (c == C_IN - 1) ? (KERNELS + 1) : KERNELS;

    // ---- B fragments (weights), hoisted: W[(m,j), n], K = v + 2*half ----
    v2f bfrag[5];
#pragma unroll
    for (int kk = 0; kk < 5; ++kk) {
#pragma unroll
        for (int v = 0; v < 2; ++v) {
            const int f = kk * 4 + v + 2 * half;   // feature index = m*3 + j
            float w = 0.0f;
            if (f < NFEAT) {
                if (n < KERNELS)                         w = conv_w[n * NFEAT + f];
                else if (n == KERNELS && c == C_IN - 1)  w = leftout_w[f];
            }
            bfrag[kk][v] = w;
        }
    }

    float bias = 0.0f;
    if (n < KERNELS)                         bias = conv_b[n];
    else if (n == KERNELS && c == C_IN - 1)  bias = leftout_b[0];

    const float* xb = x + (size_t)b * LL * C_IN + c;

    // ---- per-lane constant byte offsets for the fast path ----
    // feature f -> x position (l0 + row - 1) + (j - 3m); offset = (j-3m)*28 B
    int offb[5][2];
#pragma unroll
    for (int kk = 0; kk < 5; ++kk) {
#pragma unroll
        for (int v = 0; v < 2; ++v) {
            const int f = kk * 4 + v + 2 * half;
            const int m = f / 3;
            const int j = f - 3 * m;
            offb[kk][v] = (f < NFEAT) ? (j - 3 * m) * (int)(C_IN * sizeof(float)) : 0;
        }
    }

    for (int t = 0; t < TPW; ++t) {
        const int l0 = l0_base + t * 16;

        v8f acc;
#pragma unroll
        for (int i = 0; i < 8; ++i) acc[i] = bias;   // D = A*B + C, bias in C

        if (l0 != 0 && l0 != LL - 16) {
            // ---------- fast path: no wrap, no zero-pad ----------
            const char* pa = (const char*)(xb + (size_t)(l0 + row - 1) * C_IN);
#pragma unroll
            for (int kk = 0; kk < 4; ++kk) {         // f <= 15: always valid
                v2f afrag;
#pragma unroll
                for (int v = 0; v < 2; ++v)
                    afrag[v] = *(const float*)(pa + offb[kk][v]);
                acc = __builtin_amdgcn_wmma_f32_16x16x4_f32(
                    false, afrag, false, bfrag[kk], (short)0, acc, false, false);
            }
            {                                         // kk==4: K-pad in upper half
                v2f afrag;
#pragma unroll
                for (int v = 0; v < 2; ++v)
                    afrag[v] = (half == 0) ? *(const float*)(pa + offb[4][v]) : 0.0f;
                acc = __builtin_amdgcn_wmma_f32_16x16x4_f32(
                    false, afrag, false, bfrag[4], (short)0, acc, false, false);
            }
        } else {
            // ---------- edge path: circular wrap + front zero-pad ----------
#pragma unroll
            for (int kk = 0; kk < 5; ++kk) {
                v2f afrag;
#pragma unroll
                for (int v = 0; v < 2; ++v) {
                    const int f = kk * 4 + v + 2 * half;
                    float val = 0.0f;
                    if (f < NFEAT) {
                        const int m = f / 3;
                        const int j = f - 3 * m;
                        const int lp = (l0 + row + j - 1 + LL) & (LL - 1);
                        if (lp >= MDEL * TAO)
                            val = xb[(size_t)(lp - TAO * m) * C_IN];
                    }
                    afrag[v] = val;
                }
                acc = __builtin_amdgcn_wmma_f32_16x16x4_f32(
                    false, afrag, false, bfrag[kk], (short)0, acc, false, false);
            }
        }

        // ---- store 16x16 tile; lane = channel, vgpr = row (M = rr + 8*half) ----
        if (n < Nc) {
            const int ch = c * KERNELS + n;          // c==6,n==73 -> 511 (leftout)
            float* op = out + ((size_t)b * LL + l0 + 8 * half) * DMODEL + ch;
#pragma unroll
            for (int rr = 0; rr < 8; ++rr) {
                __builtin_nontemporal_store(acc[rr], op);  // 256MB stream, bypass L2
                op += DMODEL;
            }
        }
    }
}

extern "C" void kernel_launch(void* const* d_in, const int* in_sizes, int n_in,
                              void* d_out, int out_size, void* d_ws, size_t ws_size,
                              hipStream_t stream) {
    const float* x         = (const float*)d_in[0];
    const float* conv_w    = (const float*)d_in[1];
    const float* conv_b    = (const float*)d_in[2];
    const float* leftout_w = (const float*)d_in[3];
    const float* leftout_b = (const float*)d_in[4];
    float* out = (float*)d_out;

    dim3 block(256);                            // 8 waves / block (wave32)
    dim3 grid(XBLOCKS, C_IN * NTILE, BB);       // (4, 35, 32)
    delay_conv_wmma_kernel<<<grid, block, 0, stream>>>(
        x, conv_w, conv_b, leftout_w, leftout_b, out);
}